// Int4WeightOnlyEmbedding_14886356648419
// MI455X (gfx1250) — compile-verified
//
#include <hip/hip_runtime.h>

// ---------------------------------------------------------------------------
// Int4 weight-only embedding gather + dequant for MI455X (gfx1250).
//
//   out[r, :] = (weight[x[r], :] - zp[x[r], g]) * scale[x[r], g],  g = d/32
//
// Memory-bound: ~512 MB total traffic -> ~22 us floor at 23.3 TB/s.
// Data path: TDM (tensor_load_to_lds) DMAs each 4 KB row into LDS with
// double buffering; waves dequantize from LDS and emit non-temporal B128
// stores so the streaming output does not evict the weight table from L2.
// ---------------------------------------------------------------------------

typedef unsigned int v4u __attribute__((ext_vector_type(4)));
typedef int          v8i __attribute__((ext_vector_type(8)));
typedef int          v4i __attribute__((ext_vector_type(4)));
typedef float        v4f __attribute__((ext_vector_type(4)));

#define DIM       1024          // embedding width
#define GROUPS    32            // NG = D / 32
#define ROW_BYTES (DIM * 4)     // one int32 row = 4 KB

// Build a D# for a 1-D tile (one 4 KB weight row) and issue TENSOR_LOAD_TO_LDS.
// Group 0: count=1 | lds_addr | global_addr[56:0] | type=2
// Group 1: data_size=4B, tensor_dim0=1024, tensor_dim1=1, tile_dim0=1024,
//          tile_dim1=1, tensor_dim0_stride=1024. Remaining groups zero
//          (tile dims 2/3/4 = 0 => unused).
__device__ __forceinline__ void tdm_load_row(const int* grow, unsigned lds_off) {
    unsigned long long ga = (unsigned long long)(const void*)grow;
    v4u g0;
    g0.x = 1u;                                              // count=1, user desc
    g0.y = lds_off;                                         // LDS byte address
    g0.z = (unsigned)ga;                                    // global_addr[31:0]
    g0.w = ((unsigned)(ga >> 32) & 0x01FFFFFFu) | (2u << 30); // addr[56:32] | type=2
    v8i g1;
    g1[0] = 0x00020000;            // workgroup_mask=0, data_size=2 (4 bytes)
    g1[1] = (int)(DIM << 16);      // tensor_dim0[15:0] in bits[63:48] -> 1024
    g1[2] = (int)(1u  << 16);      // tensor_dim0 hi16 = 0, tensor_dim1 lo16 = 1
    g1[3] = (int)(DIM << 16);      // tensor_dim1 hi16 = 0, tile_dim0 = 1024
    g1[4] = 1;                     // tile_dim1 = 1, tile_dim2 = 0
    g1[5] = DIM;                   // tensor_dim0_stride lo32 = 1024
    g1[6] = 0;                     // stride0 hi16 = 0, stride1 lo16 = 0
    g1[7] = 0;                     // stride1 hi32 = 0
    v4i z4 = {0, 0, 0, 0};
    v8i z8 = {0, 0, 0, 0, 0, 0, 0, 0};
    // clang-23 / therock-10.0 form: 6 args, extra int32x8 group before cpol.
    __builtin_amdgcn_tensor_load_to_lds(g0, g1, z4, z4, z8, 0);
}

__global__ __launch_bounds__(256) void int4_embed_tdm_kernel(
    const int*   __restrict__ x,          // [R] row indices
    const int*   __restrict__ weight,     // [V, 1024] int4 values as int32
    const float* __restrict__ scale,      // [V, 32]
    const int*   __restrict__ zero_point, // [V, 32]
    float*       __restrict__ out,        // [R, 1024]
    int R)
{
    __shared__ __align__(16) int smem[2 * DIM];   // ping-pong row buffers (8 KB)

    const int  tid    = threadIdx.x;
    const bool issuer = (tid < 32);               // wave 0 drives the TDM
    const unsigned lds_base =
        (unsigned)(unsigned long long)(const void*)&smem[0]; // LDS byte offset

    // Prologue: kick off DMA for this block's first row into buffer 0.
    long r0 = (long)blockIdx.x;
    int  vcur = 0;
    if (r0 < R) {
        vcur = __builtin_amdgcn_readfirstlane(x[r0]);
        if (issuer)
            tdm_load_row(weight + (unsigned long long)vcur * DIM, lds_base);
    }

    int it = 0;
    for (long r = r0; r < R; r += (long)gridDim.x, ++it) {
        const long rn  = r + (long)gridDim.x;
        const int  cur = it & 1;

        // Issue DMA for the next row into the other buffer (its previous
        // consumers finished at the end-of-iteration barrier of it-1).
        int vnext = 0;
        if (rn < R) {
            vnext = __builtin_amdgcn_readfirstlane(x[rn]);
            if (issuer)
                tdm_load_row(weight + (unsigned long long)vnext * DIM,
                             lds_base + (unsigned)((cur ^ 1) * ROW_BYTES));
        }

        // Wave 0 waits for the current row's DMA (leaving at most the
        // next row's DMA outstanding), then releases the workgroup.
        if (issuer) {
            if (rn < R) __builtin_amdgcn_s_wait_tensorcnt(1);
            else        __builtin_amdgcn_s_wait_tensorcnt(0);
        }
        __syncthreads();

        // Dequantize 4 contiguous elements per thread (256 thr * 4 = 1024).
        const int   g    = tid >> 3;                   // group = (tid*4)/32
        const long  sidx = (long)vcur * GROUPS + g;
        const float s    = scale[sidx];
        const int   zp   = zero_point[sidx];

        const v4i w = *reinterpret_cast<const v4i*>(&smem[cur * DIM + tid * 4]);
        v4f o;
        o.x = (float)(w.x - zp) * s;
        o.y = (float)(w.y - zp) * s;
        o.z = (float)(w.z - zp) * s;
        o.w = (float)(w.w - zp) * s;

        // Streaming output: non-temporal B128 store, keep L2 for the table.
        __builtin_nontemporal_store(
            o, reinterpret_cast<v4f*>(out + r * DIM + (long)tid * 4));

        __syncthreads();   // all reads of smem[cur] done before it is re-filled
        vcur = vnext;
    }
}

extern "C" void kernel_launch(void* const* d_in, const int* in_sizes, int n_in,
                              void* d_out, int out_size, void* d_ws, size_t ws_size,
                              hipStream_t stream) {
    const int*   x  = (const int*)d_in[0];     // indices
    const int*   w  = (const int*)d_in[1];     // quantized weight
    const float* sc = (const float*)d_in[2];   // scale
    const int*   zp = (const int*)d_in[3];     // zero_point
    float*       out = (float*)d_out;

    const int R  = in_sizes[0];                // 32*2048 = 65536 rows
    int nb = R < 16384 ? R : 16384;            // 4 rows/block -> pipelined loop
    if (nb < 1) nb = 1;

    int4_embed_tdm_kernel<<<dim3(nb), dim3(256), 0, stream>>>(x, w, sc, zp, out, R);
}